// Distance_20031727469007
// MI455X (gfx1250) — compile-verified
//
#include <hip/hip_runtime.h>
#include <hip/hip_bf16.h>

typedef __attribute__((ext_vector_type(2))) float        v2f;
typedef __attribute__((ext_vector_type(8))) float        v8f;
typedef __attribute__((ext_vector_type(4))) unsigned int v4u;

#define M_PTS     128
#define KNN       32
#define PITCH     132            // u32 per d2 row (16B aligned; 2-way bank conflict worst case)
#define CUT2_BITS 0x41C80000u    // __float_as_uint(25.0f), low 7 bits zero
#define KEY_MASK  0xFFFFFF80u
#define POISON    0xFFFFFFFFu

// LDS plan (dynamic): px[128] py[128] pz[128] sq[128] keys[128*PITCH]
#define LDS_WORDS (4 * M_PTS + M_PTS * PITCH)

static __device__ __forceinline__ unsigned int umin2(unsigned int a, unsigned int b) {
    return a < b ? a : b;
}

__global__ __launch_bounds__(256) void knn_edges_kernel(
    const float* __restrict__ pos,   // (B*128, 3)
    float* __restrict__ out,         // [src(E), dst(E), w(E), vec(3E)]
    int nbatch)
{
    extern __shared__ float smem[];
    float* px = smem;                 // [0, 128)
    float* py = px + M_PTS;           // [128, 256)
    float* pz = py + M_PTS;           // [256, 384)
    float* sq = pz + M_PTS;           // [384, 512)
    unsigned int* keys = (unsigned int*)(sq + M_PTS);

    const int b   = blockIdx.x;
    const int tid = threadIdx.x;
    const float* pb = pos + (size_t)b * M_PTS * 3;

    // ---- Phase 0: stage points + squared norms into LDS ----
    if (tid < M_PTS) {
        float x = pb[tid * 3 + 0];
        float y = pb[tid * 3 + 1];
        float z = pb[tid * 3 + 2];
        px[tid] = x; py[tid] = y; pz[tid] = z;
        sq[tid] = x * x + y * y + z * z;
    }
    __syncthreads();

    // ---- Phase 1: Gram matrix via V_WMMA_F32_16X16X4_F32 ----
    // wave w owns rows [16w, 16w+16); 8 column tiles of 16, fully unrolled
    // so the 8 independent WMMAs pipeline (no D->A/B RAW between them).
    const int  wave  = tid >> 5;          // 0..7
    const int  lane  = tid & 31;
    const int  lhalf = lane & 15;
    const bool hi    = (lane >= 16);
    const int  r0    = wave * 16;
    const int  mrow  = r0 + lhalf;
    const int  mbase = hi ? 8 : 0;

    // Branchless fragment build: address select + multiply masks (EXEC stays all-ones).
    // A 16x4 f32: lanes 0-15 carry K={0,1}; lanes 16-31 carry K={2,3} (K=3 pad=0).
    // Bake -2 into A so D = -2 * P.P^T.
    const int   xsel = hi ? 2 * M_PTS : 0;   // px (lo half) or pz (hi half)
    const float amul = hi ? 0.0f : -2.0f;    // y-slot: -2*py (lo) / zero pad (hi)
    const float bmul = hi ? 0.0f :  1.0f;

    v2f afrag;
    afrag.x = -2.0f * smem[xsel + mrow];
    afrag.y = amul  * py[mrow];

    float sqrow[8];
#pragma unroll
    for (int r = 0; r < 8; ++r)
        sqrow[r] = sq[r0 + mbase + r];

#pragma unroll
    for (int jt = 0; jt < 8; ++jt) {
        const int c0   = jt * 16;
        const int ncol = c0 + lhalf;

        v2f bfrag;
        bfrag.x = smem[xsel + ncol];
        bfrag.y = bmul * py[ncol];

        v8f acc = {};
        acc = __builtin_amdgcn_wmma_f32_16x16x4_f32(
            /*neg_a=*/false, afrag, /*neg_b=*/false, bfrag,
            /*c_mod=*/(short)0, acc, /*reuse_a=*/false, /*reuse_b=*/false);

        const float sqn = sq[ncol];
#pragma unroll
        for (int r = 0; r < 8; ++r) {
            // C/D layout: VGPR r -> row mbase+r, lane%16 -> column
            float v = acc[r] + sqrow[r] + sqn;
            v = fmaxf(v, 0.0f);
            // Pack sort key: monotonic positive-float bits | 7-bit column index.
            // u32-min then selects smallest d2, ties -> lowest column (top_k rule),
            // and key < bits(25.0f) is an EXACT d2 < 25.0 test (mask only rounds down).
            unsigned int kv = (__float_as_uint(v) & KEY_MASK) | (unsigned int)ncol;
            keys[(r0 + mbase + r) * PITCH + ncol] = kv;
        }
    }
    __syncthreads();

    // ---- Phase 2: per-row top-K=32 by iterative u32-min extraction ----
    const size_t E = (size_t)nbatch * M_PTS * KNN;
    if (tid < M_PTS) {
        const int row       = tid;
        unsigned int* drow  = keys + row * PITCH;
        const float ptx = px[row], pty = py[row], ptz = pz[row];
        const int    node_base = b * M_PTS;
        const size_t ebase     = ((size_t)b * M_PTS + row) * KNN;

        float* src_o = out;
        float* dst_o = out + E;
        float* w_o   = out + 2 * E;
        float* v_o   = out + 3 * E;

        const float dstv = (float)(node_base + row);

#pragma unroll 1
        for (int k = 0; k < KNN; ++k) {
            unsigned int mk = POISON;
#pragma unroll 8
            for (int j4 = 0; j4 < M_PTS / 4; ++j4) {
                v4u q = *(const v4u*)(drow + j4 * 4);
                unsigned int m01 = umin2(q.x, q.y);
                unsigned int m23 = umin2(q.z, q.w);
                mk = umin2(mk, umin2(m01, m23));
            }

            const bool taken = (mk < CUT2_BITS);   // exact d2 < 25.0 test
            int   idx;
            float wgt, vx, vy, vz;
            if (taken) {
                idx = (int)(mk & 127u);
                drow[idx] = POISON;                // remove from future rounds
                vx = px[idx] - ptx;
                vy = py[idx] - pty;
                vz = pz[idx] - ptz;
                const float sqv = vx * vx + vy * vy + vz * vz;
                wgt = (idx != row) ? sqrtf(sqv) : 0.0f;
            } else {
                idx = row;                         // self fallback, weight/vec zero
                wgt = 0.0f; vx = 0.0f; vy = 0.0f; vz = 0.0f;
            }

            const size_t e = ebase + k;
            __builtin_nontemporal_store((float)(node_base + idx), src_o + e);
            __builtin_nontemporal_store(dstv,                     dst_o + e);
            __builtin_nontemporal_store(wgt,                      w_o + e);
            __builtin_nontemporal_store(vx,                       v_o + 3 * e + 0);
            __builtin_nontemporal_store(vy,                       v_o + 3 * e + 1);
            __builtin_nontemporal_store(vz,                       v_o + 3 * e + 2);
        }
    }
}

extern "C" void kernel_launch(void* const* d_in, const int* in_sizes, int n_in,
                              void* d_out, int out_size, void* d_ws, size_t ws_size,
                              hipStream_t stream) {
    const float* pos = (const float*)d_in[0];
    (void)n_in; (void)d_ws; (void)ws_size; (void)out_size;

    const int nbatch = in_sizes[0] / (M_PTS * 3);   // B = 1024 for the reference shapes
    const size_t lds_bytes = (size_t)LDS_WORDS * sizeof(float);  // ~68 KB < 320 KB WGP LDS

    knn_edges_kernel<<<dim3(nbatch), dim3(256), lds_bytes, stream>>>(
        pos, (float*)d_out, nbatch);
}